// Att_cat_withid_norm_42984032698917
// MI455X (gfx1250) — compile-verified
//
#include <hip/hip_runtime.h>
#include <math.h>

typedef __attribute__((ext_vector_type(2))) float v2f;
typedef __attribute__((ext_vector_type(4))) float v4f;
typedef __attribute__((ext_vector_type(8))) float v8f;

#define NROW 1024   // B*N
#define NPB  512    // N
#define DD   64     // D

// ---------------------------------------------------------------------------
// Kernel A: LayerNorm rows of ua, then scores via V_WMMA_F32_16X16X4_F32.
//   S[1024 x 16] = LN(ua)[1024 x 64] @ W[64 x 16]   (col0 = w_q, col1 = w_k)
// Each wave handles 16 rows; 8 waves/block; 8 blocks -> 1024 rows.
// Also computes c[b] = LN(iid[b]) . w[128:192] + b_att.
// ---------------------------------------------------------------------------
__global__ void __launch_bounds__(256)
score_kernel(const float* __restrict__ ua, const float* __restrict__ iid,
             const float* __restrict__ ln_g, const float* __restrict__ ln_b,
             const float* __restrict__ w_att, const float* __restrict__ b_att,
             float* __restrict__ sq, float* __restrict__ sk, float* __restrict__ cb)
{
    constexpr int LST = 68;                 // LDS row stride (bank-conflict free)
    __shared__ float lnbuf[8][16 * LST];
    const int tid  = threadIdx.x;
    const int wv   = tid >> 5;
    const int lane = tid & 31;
    const int rowBase = blockIdx.x * 128 + wv * 16;

    // ---- LayerNorm: lanes 0..15 each normalize one row of 64 into LDS ----
    if (lane < 16) {
        const v4f* src = (const v4f*)(ua + (size_t)(rowBase + lane) * DD);
        float s = 0.f;
        for (int q = 0; q < 16; ++q) { v4f t = src[q]; s += t.x + t.y + t.z + t.w; }
        const float mean = s * (1.f / 64.f);
        float var = 0.f;
        for (int q = 0; q < 16; ++q) {
            v4f t = src[q];
            float a0 = t.x - mean, a1 = t.y - mean, a2 = t.z - mean, a3 = t.w - mean;
            var += a0 * a0 + a1 * a1 + a2 * a2 + a3 * a3;
        }
        const float rstd = rsqrtf(var * (1.f / 64.f) + 1e-5f);
        for (int q = 0; q < 16; ++q) {
            v4f t = src[q];
            v4f g = ((const v4f*)ln_g)[q];
            v4f o = ((const v4f*)ln_b)[q];
            float* dst = &lnbuf[wv][lane * LST + 4 * q];
            dst[0] = (t.x - mean) * rstd * g.x + o.x;
            dst[1] = (t.y - mean) * rstd * g.y + o.y;
            dst[2] = (t.z - mean) * rstd * g.z + o.z;
            dst[3] = (t.w - mean) * rstd * g.w + o.w;
        }
    }
    __syncthreads();

    // ---- WMMA: A = LN tile [16x4], B = W tile [4x16] (cols 0,1 nonzero) ----
    // A f32 layout: VGPR0 -> K = 2*half, VGPR1 -> K = 2*half+1 ; M = lane&15.
    // B operand built branch-free: unconditional load + multiplicative mask so
    // the 16 WMMAs issue back-to-back without exec-mask churn.
    const int half = lane >> 4;
    const int lm   = lane & 15;
    const float wmask = (lm < 2) ? 1.f : 0.f;
    const float* wc   = w_att + (size_t)(lm & 1) * 64;
    v8f acc = {};
#pragma unroll
    for (int k = 0; k < 64; k += 4) {
        const int kk = k + 2 * half;
        v2f av, bv;
        av.x = lnbuf[wv][lm * LST + kk];
        av.y = lnbuf[wv][lm * LST + kk + 1];
        bv.x = wc[kk]     * wmask;
        bv.y = wc[kk + 1] * wmask;
        acc = __builtin_amdgcn_wmma_f32_16x16x4_f32(false, av, false, bv,
                                                    (short)0, acc, false, false);
    }

    // C/D layout: VGPR v, lanes 0-15 -> M=v, lanes 16-31 -> M=v+8; N = lane&15.
    if (lm < 2) {
        float* dst = (lm == 0) ? sq : sk;
#pragma unroll
        for (int v = 0; v < 8; ++v) dst[rowBase + 8 * half + v] = acc[v];
    }

    // ---- per-batch constant: LN(iid[b]) . w[128:192] + b_att ----
    if (blockIdx.x == 0 && tid < 2) {
        const float* x = iid + tid * DD;
        float s = 0.f;
        for (int d = 0; d < DD; ++d) s += x[d];
        const float mean = s * (1.f / 64.f);
        float var = 0.f;
        for (int d = 0; d < DD; ++d) { float t = x[d] - mean; var += t * t; }
        const float rstd = rsqrtf(var * (1.f / 64.f) + 1e-5f);
        float dot = 0.f;
        for (int d = 0; d < DD; ++d)
            dot += ((x[d] - mean) * rstd * ln_g[d] + ln_b[d]) * w_att[128 + d];
        cb[tid] = dot + b_att[0];
    }
}

// ---------------------------------------------------------------------------
// Kernel B: per row (b,i), softmax stats over j:
//   m  = max_j lrelu(sq_i + sk_j + c_b),  rd = 1 / sum_j exp(score - m)
// One wave32 per row; shuffle-tree reductions.
// ---------------------------------------------------------------------------
__global__ void __launch_bounds__(256)
stats_kernel(const float* __restrict__ sq, const float* __restrict__ sk,
             const float* __restrict__ cb, float* __restrict__ mrow,
             float* __restrict__ rdrow)
{
    const int tid  = threadIdx.x;
    const int row  = blockIdx.x * 8 + (tid >> 5);
    const int lane = tid & 31;
    const int b    = row >> 9;
    const float base = sq[row] + cb[b];
    const float* skb = sk + (b << 9);

    float mx = -3.402823466e38f;
    for (int j = lane; j < NPB; j += 32) {
        float s = base + skb[j];
        s = (s > 0.f) ? s : 0.01f * s;
        mx = fmaxf(mx, s);
    }
    for (int o = 16; o > 0; o >>= 1) mx = fmaxf(mx, __shfl_xor(mx, o, 32));

    float sum = 0.f;
    for (int j = lane; j < NPB; j += 32) {
        float s = base + skb[j];
        s = (s > 0.f) ? s : 0.01f * s;
        sum += __expf(s - mx);
    }
    for (int o = 16; o > 0; o >>= 1) sum += __shfl_xor(sum, o, 32);

    if (lane == 0) { mrow[row] = mx; rdrow[row] = 1.f / sum; }
}

// ---------------------------------------------------------------------------
// Kernel C: the 256 MiB writer (bandwidth bound). One block per (b,i) row.
//   alphas[b,i,j,d] = exp(lrelu(sq_i+sk_j+c_b)-m)/sum   (broadcast over d)
//   value [b,i,j,d] = ua[b,i,d] * ua[b,j,d]
// 16 threads per j x float4 over d -> fully coalesced 128-bit NT stores.
// ---------------------------------------------------------------------------
__global__ void __launch_bounds__(256)
writer_kernel(const float* __restrict__ ua, const float* __restrict__ sq,
              const float* __restrict__ sk, const float* __restrict__ cb,
              const float* __restrict__ mrow, const float* __restrict__ rdrow,
              float* __restrict__ out)
{
    const int row = blockIdx.x;            // b*512 + i
    const int b   = row >> 9;
    const int tid = threadIdx.x;
    const int dg  = tid & 15;              // float4 group within D=64
    const int jl  = tid >> 4;              // 0..15

    const v4f* ua4 = (const v4f*)ua;
    const v4f  qv  = ua4[(size_t)row * 16 + dg];
    const float base = sq[row] + cb[b];
    const float mi   = mrow[row];
    const float rdi  = rdrow[row];
    const v4f* kb    = ua4 + ((size_t)(b << 9)) * 16;
    const float* skb = sk + (b << 9);

    v4f* aout = (v4f*)out;
    v4f* vout = aout + (size_t)NROW * NPB * 16;   // value follows alphas
    const size_t outBase = (size_t)row * NPB * 16;

    for (int jt = 0; jt < 32; ++jt) {
        const int j = (jt << 4) + jl;
        v4f kv = kb[(size_t)j * 16 + dg];
        float s = base + skb[j];
        s = (s > 0.f) ? s : 0.01f * s;
        const float a = __expf(s - mi) * rdi;
        const size_t idx = outBase + (size_t)j * 16 + dg;
        v4f av = {a, a, a, a};
        v4f vv = qv * kv;
        __builtin_nontemporal_store(av, aout + idx);
        __builtin_nontemporal_store(vv, vout + idx);
    }
}

// ---------------------------------------------------------------------------
extern "C" void kernel_launch(void* const* d_in, const int* in_sizes, int n_in,
                              void* d_out, int out_size, void* d_ws, size_t ws_size,
                              hipStream_t stream) {
    const float* ua    = (const float*)d_in[0];
    const float* iid   = (const float*)d_in[1];
    const float* ln_g  = (const float*)d_in[2];
    const float* ln_b  = (const float*)d_in[3];
    const float* w_att = (const float*)d_in[4];
    const float* b_att = (const float*)d_in[5];

    float* ws    = (float*)d_ws;
    float* sq    = ws;              // 1024
    float* sk    = ws + 1024;       // 1024
    float* cb    = ws + 2048;       // 2
    float* mrow  = ws + 2064;       // 1024
    float* rdrow = ws + 3088;       // 1024

    score_kernel <<<8,    256, 0, stream>>>(ua, iid, ln_g, ln_b, w_att, b_att, sq, sk, cb);
    stats_kernel <<<128,  256, 0, stream>>>(sq, sk, cb, mrow, rdrow);
    writer_kernel<<<1024, 256, 0, stream>>>(ua, sq, sk, cb, mrow, rdrow, (float*)d_out);
}